// Net_549755814512
// MI455X (gfx1250) — compile-verified
//
#include <hip/hip_runtime.h>

typedef __attribute__((ext_vector_type(2))) float v2f;
typedef __attribute__((ext_vector_type(4))) float v4f;
typedef __attribute__((ext_vector_type(8))) float v8f;

// ---------------------------------------------------------------------------
// Kernel 1: fold the whole binarized net into a 96-float effective weight
// vector + scalar bias, and compute the two quantization-error scalars.
// Runs once on a single block; cost is negligible (~10k ops).
//   ws[0..95]  = Weff, ws[96] = bias_eff
//   err_out[0] = error1, err_out[1] = error2   (err_out = d_out + B)
// ---------------------------------------------------------------------------
__global__ void prep_kernel(const float* __restrict__ W1, const float* __restrict__ b1,
                            const float* __restrict__ W2, const float* __restrict__ b2,
                            float* __restrict__ ws, float* __restrict__ err_out) {
    __shared__ float sW1[96], sWb1[96], sW2[100], sWb2[100];
    __shared__ float sAlpha1[4], sAlpha2;
    const int t = threadIdx.x;

    if (t < 96)  sW1[t] = W1[t];
    if (t < 100) sW2[t] = W2[t];
    __syncthreads();

    if (t < 4) {                       // alpha1[c] = mean |W1[c,:,:,:]| over 24
        float s = 0.f;
        for (int i = 0; i < 24; ++i) s += fabsf(sW1[t * 24 + i]);
        sAlpha1[t] = s * (1.0f / 24.0f);
    }
    if (t == 4) {                      // alpha2 = mean |W2| over 100
        float s = 0.f;
        for (int i = 0; i < 100; ++i) s += fabsf(sW2[i]);
        sAlpha2 = s * 0.01f;
    }
    __syncthreads();

    if (t < 96) {                      // Wb1 = alpha1 * sign(W1), sign(0)=+1
        float a = sAlpha1[t / 24];
        sWb1[t] = (sW1[t] >= 0.f) ? a : -a;
    }
    if (t < 100) {
        float a = sAlpha2;
        sWb2[t] = (sW2[t] >= 0.f) ? a : -a;
    }
    __syncthreads();

    if (t == 0) {
        float e1 = 0.f;
        for (int i = 0; i < 96; ++i) { float d = sW1[i] - sWb1[i]; e1 += d * d; }
        err_out[0] = e1 * (1.0f / 96.0f);
        float e2 = 0.f;
        for (int i = 0; i < 100; ++i) { float d = sW2[i] - sWb2[i]; e2 += d * d; }
        err_out[1] = e2 * 0.01f;
        // bias_eff = b2 + sum_c b1[c] * sum_j Wb2[c*25+j]
        float bias = b2[0];
        for (int c = 0; c < 4; ++c) {
            float s2 = 0.f;
            for (int j = 0; j < 25; ++j) s2 += sWb2[c * 25 + j];
            bias += b1[c] * s2;
        }
        ws[96] = bias;
    }
    if (t < 96) {                      // Weff[kh*32+w] (x flat layout is b*96 + kh*32 + w)
        const int kh = t >> 5, w = t & 31;
        int j0 = w - 7; if (j0 < 0) j0 = 0;
        int j1 = w;     if (j1 > 24) j1 = 24;
        float s = 0.f;
        for (int c = 0; c < 4; ++c)
            for (int j = j0; j <= j1; ++j)
                s += sWb2[c * 25 + j] * sWb1[c * 24 + kh * 8 + (w - j)];
        ws[t] = s;
    }
}

// ---------------------------------------------------------------------------
// Kernel 2: streaming GEMV  y = X(B x 96) . Weff + bias  via V_WMMA_F32_16X16X4_F32.
// One wave per 16-row tile (grid-stride). A is loaded directly in the ISA's
// 16x4 f32 layout (lanes 0-15: K=0/1, lanes 16-31: K=2/3, M = lane%16) with
// per-lane b64 loads; B holds Weff replicated across N. 24 chained WMMAs
// reduce K=96; column N=0 (lanes 0 & 16) carries the 16 results.
// ---------------------------------------------------------------------------
__global__ void gemv_wmma(const float* __restrict__ x, const float* __restrict__ ws,
                          float* __restrict__ y, int B) {
    const int lane          = threadIdx.x & 31;
    const int waveInBlock   = threadIdx.x >> 5;
    const int wavesPerBlock = blockDim.x >> 5;
    const long waveId     = (long)blockIdx.x * wavesPerBlock + waveInBlock;
    const long waveStride = (long)gridDim.x * wavesPerBlock;
    const int  nTiles     = B >> 4;

    const int koff = (lane >> 4) << 1;            // 0 for lanes 0-15, 2 for lanes 16-31

    // B-matrix chunks (Weff replicated over N), loaded once per wave.
    v2f bvec[24];
#pragma unroll
    for (int kk = 0; kk < 24; ++kk)
        bvec[kk] = *(const v2f*)(ws + 4 * kk + koff);
    const float biasEff = ws[96];

    const int  rowInTile  = lane & 15;
    const long laneOffset = (long)rowInTile * 96 + koff;   // floats

    for (long tile = waveId; tile < nTiles; tile += waveStride) {
        const float* xt = x + tile * (16 * 96) + laneOffset;
        v8f c = {0.f, 0.f, 0.f, 0.f, 0.f, 0.f, 0.f, 0.f};
#pragma unroll
        for (int kk = 0; kk < 24; ++kk) {
            v2f a = *(const v2f*)(xt + 4 * kk);
            // D = A(16x4 f32) * B(4x16 f32) + C
            c = __builtin_amdgcn_wmma_f32_16x16x4_f32(
                    false, a, false, bvec[kk], (short)0, c, false, false);
        }
        // Lane 0 holds (M=g, N=0) in c[g]; lane 16 holds (M=8+g, N=0).
        if ((lane & 15) == 0) {
            float* yo = y + tile * 16 + ((lane >> 4) << 3);
            v4f lo = {c[0] + biasEff, c[1] + biasEff, c[2] + biasEff, c[3] + biasEff};
            v4f hi = {c[4] + biasEff, c[5] + biasEff, c[6] + biasEff, c[7] + biasEff};
            *(v4f*)(yo)     = lo;
            *(v4f*)(yo + 4) = hi;
        }
    }

    // Tail rows (B % 16) — scalar dot, first wave only (no-op for B=524288).
    if (waveId == 0) {
        for (int r = (nTiles << 4) + lane; r < B; r += 32) {
            float acc = biasEff;
            const float* xr = x + (long)r * 96;
            for (int i = 0; i < 96; ++i) acc += xr[i] * ws[i];
            y[r] = acc;
        }
    }
}

extern "C" void kernel_launch(void* const* d_in, const int* in_sizes, int n_in,
                              void* d_out, int out_size, void* d_ws, size_t ws_size,
                              hipStream_t stream) {
    const float* x  = (const float*)d_in[0];
    const float* W1 = (const float*)d_in[1];
    const float* b1 = (const float*)d_in[2];
    const float* W2 = (const float*)d_in[3];
    const float* b2 = (const float*)d_in[4];
    float* out = (float*)d_out;
    float* ws  = (float*)d_ws;

    const int B = in_sizes[0] / 96;          // x is (B,1,3,32)

    // Fold the net into Weff/bias_eff and emit the two error scalars.
    prep_kernel<<<1, 128, 0, stream>>>(W1, b1, W2, b2, ws, out + B);

    // Streaming WMMA GEMV over 201 MB of x: bandwidth-bound (~8.7 us at 23.3 TB/s).
    const int threads = 256;                 // 8 waves/block (wave32)
    const int wavesPerBlock = threads / 32;
    const int nTiles = B / 16;
    int blocks = 2048;                       // grid-stride; ~2 tiles per wave at B=524288
    if ((long)blocks * wavesPerBlock > nTiles)
        blocks = (nTiles + wavesPerBlock - 1) / wavesPerBlock;
    if (blocks < 1) blocks = 1;
    gemv_wmma<<<blocks, threads, 0, stream>>>(x, ws, out, B);
}